// GraphAttentionModule_51711406244125
// MI455X (gfx1250) — compile-verified
//
#include <hip/hip_runtime.h>

typedef __attribute__((ext_vector_type(2))) float v2f;
typedef __attribute__((ext_vector_type(8))) float v8f;

#define IN_C   256
#define OUTF   256   // HEADS * OUT_C
#define HEADS  4
#define OUT_C  64
#define NEG_SLOPE 0.2f

// ---------------------------------------------------------------------------
// float atomic max via sign-aware integer atomics (monotonic bit trick)
// ---------------------------------------------------------------------------
__device__ __forceinline__ void atomicMaxF(float* addr, float val) {
    if (val >= 0.0f)
        atomicMax((int*)addr, __float_as_int(val));
    else
        atomicMin((unsigned int*)addr, __float_as_uint(val));
}

// ---------------------------------------------------------------------------
// Kernel 1: h = x @ W   (fp32 WMMA 16x16x4, one wave -> 16x64 output strip)
//
// A (16x4 f32) lane layout:  m = lane%16 ; VGPR0/1 hold K = 2*(lane/16)+{0,1}
// B (4x16 f32) lane layout:  n = lane%16 ; VGPR0/1 hold K = 2*(lane/16)+{0,1}
// C/D (16x16 f32): VGPR i -> row i (lanes 0-15) / row i+8 (lanes 16-31)
// ---------------------------------------------------------------------------
__global__ __launch_bounds__(256) void k_gemm(const float* __restrict__ x,
                                              const float* __restrict__ W,
                                              float* __restrict__ h,
                                              int n, int mtiles) {
    int wid  = (blockIdx.x * blockDim.x + threadIdx.x) >> 5;
    int lane = threadIdx.x & 31;
    int tm = wid >> 2;     // 16-row tile index
    int tq = wid & 3;      // 64-column quad index
    if (tm >= mtiles) return;              // wave-uniform: EXEC stays all-ones

    int mrow = tm * 16 + (lane & 15);
    if (mrow >= n) mrow = n - 1;           // clamp partial-tile A reads
    int kgrp = (lane >> 4) << 1;           // 0 or 2
    int n0   = tq * 64 + (lane & 15);

    v8f acc0 = {}, acc1 = {}, acc2 = {}, acc3 = {};
    for (int k = 0; k < IN_C; k += 4) {
        v2f a = *(const v2f*)(x + (size_t)mrow * IN_C + k + kgrp);
        const float* wp = W + (size_t)(k + kgrp) * OUTF + n0;
        v2f b0, b1, b2, b3;
        b0.x = wp[0];  b0.y = wp[OUTF + 0];
        b1.x = wp[16]; b1.y = wp[OUTF + 16];
        b2.x = wp[32]; b2.y = wp[OUTF + 32];
        b3.x = wp[48]; b3.y = wp[OUTF + 48];
        acc0 = __builtin_amdgcn_wmma_f32_16x16x4_f32(false, a, false, b0, (short)0, acc0, false, false);
        acc1 = __builtin_amdgcn_wmma_f32_16x16x4_f32(false, a, false, b1, (short)0, acc1, false, false);
        acc2 = __builtin_amdgcn_wmma_f32_16x16x4_f32(false, a, false, b2, (short)0, acc2, false, false);
        acc3 = __builtin_amdgcn_wmma_f32_16x16x4_f32(false, a, false, b3, (short)0, acc3, false, false);
    }

    int row0 = tm * 16 + ((lane >> 4) << 3);
    int colb = tq * 64 + (lane & 15);
#pragma unroll
    for (int i = 0; i < 8; ++i) {
        int r = row0 + i;
        if (r < n) {
            float* hp = h + (size_t)r * OUTF + colb;
            hp[0]  = acc0[i];
            hp[16] = acc1[i];
            hp[32] = acc2[i];
            hp[48] = acc3[i];
        }
    }
}

// ---------------------------------------------------------------------------
// Kernel 2: per-node attention dot products   a_src/a_dst [N, H]
// ---------------------------------------------------------------------------
__global__ void k_att(const float* __restrict__ h,
                      const float* __restrict__ att_src,
                      const float* __restrict__ att_dst,
                      float* __restrict__ a_src, float* __restrict__ a_dst, int n) {
    int t = blockIdx.x * blockDim.x + threadIdx.x;
    if (t >= n * HEADS) return;
    int node = t >> 2, hd = t & 3;
    const float* hp = h + (size_t)node * OUTF + hd * OUT_C;
    const float* as = att_src + hd * OUT_C;
    const float* ad = att_dst + hd * OUT_C;
    float ss = 0.0f, sd = 0.0f;
#pragma unroll 4
    for (int c = 0; c < OUT_C; ++c) { float v = hp[c]; ss += v * as[c]; sd += v * ad[c]; }
    a_src[t] = ss;
    a_dst[t] = sd;
}

// ---------------------------------------------------------------------------
// Kernel 3a: init amax = -inf, denom = 0
// ---------------------------------------------------------------------------
__global__ void k_init_nodes(float* __restrict__ amax, float* __restrict__ denom, int n) {
    int t = blockIdx.x * blockDim.x + threadIdx.x;
    if (t >= n * HEADS) return;
    amax[t]  = -__builtin_inff();
    denom[t] = 0.0f;
}

// ---------------------------------------------------------------------------
// Kernel 3b: init out = broadcast(bias)   (bias add comes for free)
// ---------------------------------------------------------------------------
__global__ void k_init_out(const float* __restrict__ bias, float* __restrict__ out, int n) {
    int t = blockIdx.x * blockDim.x + threadIdx.x;
    if (t >= n * OUTF) return;
    out[t] = bias[t & (OUTF - 1)];
}

// ---------------------------------------------------------------------------
// Kernel 4: per-(edge,head) logits + leaky relu + segment max
// ---------------------------------------------------------------------------
__global__ void k_edge_logit(const int* __restrict__ ei,
                             const float* __restrict__ a_src,
                             const float* __restrict__ a_dst,
                             float* __restrict__ alpha, float* __restrict__ amax,
                             int n, int E) {
    int t = blockIdx.x * blockDim.x + threadIdx.x;
    int etot = E + n;
    if (t >= etot * HEADS) return;
    int e = t >> 2, hd = t & 3;
    int src, dst;
    if (e < E) { src = ei[e]; dst = ei[E + e]; }
    else       { src = dst = e - E; }           // self loops
    float a = a_src[(size_t)src * HEADS + hd] + a_dst[(size_t)dst * HEADS + hd];
    a = (a >= 0.0f) ? a : NEG_SLOPE * a;
    alpha[t] = a;
    atomicMaxF(&amax[(size_t)dst * HEADS + hd], a);
}

// ---------------------------------------------------------------------------
// Kernel 5: ex = exp(alpha - amax[dst]); denom[dst] += ex  (ex written in-place)
// ---------------------------------------------------------------------------
__global__ void k_edge_exp(const int* __restrict__ ei,
                           float* __restrict__ alpha,
                           const float* __restrict__ amax,
                           float* __restrict__ denom, int n, int E) {
    int t = blockIdx.x * blockDim.x + threadIdx.x;
    int etot = E + n;
    if (t >= etot * HEADS) return;
    int e = t >> 2, hd = t & 3;
    int dst = (e < E) ? ei[E + e] : (e - E);
    float ex = __expf(alpha[t] - amax[(size_t)dst * HEADS + hd]);
    alpha[t] = ex;
    atomicAdd(&denom[(size_t)dst * HEADS + hd], ex);
}

// ---------------------------------------------------------------------------
// Kernel 6: one wave32 per edge: out[dst] += h[src] * coef  (coalesced strips;
// head index c>>6 == j>>1 is compile-time per unrolled step)
// ---------------------------------------------------------------------------
__global__ __launch_bounds__(256) void k_scatter(const float* __restrict__ h,
                                                 const int* __restrict__ ei,
                                                 const float* __restrict__ exbuf,
                                                 const float* __restrict__ denom,
                                                 float* __restrict__ out,
                                                 int n, int E) {
    int wid  = (blockIdx.x * blockDim.x + threadIdx.x) >> 5;
    int lane = threadIdx.x & 31;
    int etot = E + n;
    if (wid >= etot) return;
    int src, dst;
    if (wid < E) { src = ei[wid]; dst = ei[E + wid]; }
    else         { src = dst = wid - E; }

    float coef[HEADS];
#pragma unroll
    for (int hd = 0; hd < HEADS; ++hd) {
        float ex = exbuf[(size_t)wid * HEADS + hd];
        float dn = denom[(size_t)dst * HEADS + hd];
        coef[hd] = ex / (dn + 1e-16f);
    }

    const float* hs = h + (size_t)src * OUTF;
    float*       op = out + (size_t)dst * OUTF;
#pragma unroll
    for (int j = 0; j < 8; ++j) {
        int c = j * 32 + lane;
        atomicAdd(&op[c], hs[c] * coef[j >> 1]);
    }
}

// ---------------------------------------------------------------------------
// Host launcher
// ---------------------------------------------------------------------------
extern "C" void kernel_launch(void* const* d_in, const int* in_sizes, int n_in,
                              void* d_out, int out_size, void* d_ws, size_t ws_size,
                              hipStream_t stream) {
    const float* x       = (const float*)d_in[0];
    const int*   ei      = (const int*)  d_in[1];   // [2, E] row-major
    const float* W       = (const float*)d_in[2];
    const float* att_src = (const float*)d_in[3];
    const float* att_dst = (const float*)d_in[4];
    const float* bias    = (const float*)d_in[5];
    float*       out     = (float*)d_out;

    const int n = in_sizes[0] / IN_C;   // 50000
    const int E = in_sizes[1] / 2;      // 800000
    const int etot = E + n;

    // workspace carve-up (floats)
    float* ws     = (float*)d_ws;
    float* h      = ws;                 size_t off  = (size_t)n * OUTF;
    float* a_src  = ws + off;           off += (size_t)n * HEADS;
    float* a_dst  = ws + off;           off += (size_t)n * HEADS;
    float* alpha  = ws + off;           off += (size_t)etot * HEADS;
    float* amax   = ws + off;           off += (size_t)n * HEADS;
    float* denom  = ws + off;           off += (size_t)n * HEADS;
    (void)ws_size; (void)n_in; (void)out_size;

    const int BLK = 256;

    // 1) GEMM: h = x @ W  (waves = mtiles * 4)
    int mtiles = (n + 15) / 16;
    int gemm_waves = mtiles * 4;
    int gemm_blocks = (gemm_waves * 32 + BLK - 1) / BLK;
    k_gemm<<<gemm_blocks, BLK, 0, stream>>>(x, W, h, n, mtiles);

    // 2) attention dot products
    k_att<<<(n * HEADS + BLK - 1) / BLK, BLK, 0, stream>>>(h, att_src, att_dst, a_src, a_dst, n);

    // 3) init reductions + output(=bias)
    k_init_nodes<<<(n * HEADS + BLK - 1) / BLK, BLK, 0, stream>>>(amax, denom, n);
    k_init_out<<<(n * OUTF + BLK - 1) / BLK, BLK, 0, stream>>>(bias, out, n);

    // 4) logits + segment max
    k_edge_logit<<<(etot * HEADS + BLK - 1) / BLK, BLK, 0, stream>>>(ei, a_src, a_dst, alpha, amax, n, E);

    // 5) exp + segment sum
    k_edge_exp<<<(etot * HEADS + BLK - 1) / BLK, BLK, 0, stream>>>(ei, alpha, amax, denom, n, E);

    // 6) weighted scatter-add (one wave per edge)
    int scat_blocks = (etot * 32 + BLK - 1) / BLK;
    k_scatter<<<scat_blocks, BLK, 0, stream>>>(h, ei, alpha, denom, out, n, E);
}